// YOLOLoss_64725157150750
// MI455X (gfx1250) — compile-verified
//
#include <hip/hip_runtime.h>
#include <hip/hip_bf16.h>

// YOLO head decode for MI455X (gfx1250).
// HBM-bound transpose+elementwise (~353 MB stream => ~15us at 23.3 TB/s).
// CDNA5 path: Tensor Data Mover DMAs each strided (85 x 52 float, row stride
// 2704 floats) channel tile into LDS; two tiles per block are pipelined
// (issue both, s_wait_tensorcnt 1 / 0) so the second DMA overlaps compute of
// the first. Output is written as contiguous coalesced non-temporal stores.

typedef unsigned int u32;
typedef unsigned long long u64;
typedef u32 v4u __attribute__((ext_vector_type(4)));
typedef int  v8i __attribute__((ext_vector_type(8)));
typedef int  v4i __attribute__((ext_vector_type(4)));

#define BS        64
#define NA        3
#define NATTR     85                  // 5 + 80 classes
#define GDIM      52
#define ROWS      2                   // y-rows (tiles) per block, double-buffered
#define TILE_ELEMS (NATTR * GDIM)     // 4420 floats = 17,680 bytes per tile
#define NBLOCKS   (BS * NA * (GDIM / ROWS))   // 4992

__device__ __forceinline__ float fast_sigmoid(float v) {
    // 1 / (1 + e^-v); __expf lowers to v_exp_f32
    return 1.0f / (1.0f + __expf(-v));
}

// Issue one TDM descriptor: 85 rows x 52 floats, row stride 2704 floats.
__device__ __forceinline__ void tdm_load_tile(const float* __restrict__ gptr, u32 lds_off) {
    const u64 gaddr = (u64)(uintptr_t)gptr;

    // D# group 0 (128b): count=1 | lds_addr | global_addr[56:0] | type=2
    v4u g0;
    g0.x = 1u;                                               // count=1, user mode
    g0.y = lds_off;                                          // lds_addr (bytes)
    g0.z = (u32)(gaddr & 0xFFFFFFFFull);                     // global_addr[31:0]
    g0.w = (u32)((gaddr >> 32) & 0x1FFFFFFull) | (2u << 30); // addr[56:32] | type=2

    // D# group 1 (256b)
    v8i g1;
    g1[0] = (int)(2u << 16);            // wg_mask=0, data_size=2 (4 bytes), no flags
    g1[1] = (int)((u32)GDIM << 16);     // atomic_barrier_addr=0 ; tensor_dim0[15:0]=52
    g1[2] = (int)((u32)NATTR << 16);    // tensor_dim0[31:16]=0 ; tensor_dim1[15:0]=85
    g1[3] = (int)((u32)GDIM << 16);     // tensor_dim1[31:16]=0 ; tile_dim0=52
    g1[4] = (int)(u32)NATTR;            // tile_dim1=85 ; tile_dim2=0 (unused)
    g1[5] = (int)(u32)(GDIM * GDIM);    // tensor_dim0_stride = 2704 (data_size units)
    g1[6] = 0;                          // stride hi ; tensor_dim1_stride lo
    g1[7] = 0;                          // tensor_dim1_stride hi

    v4i g2 = {0, 0, 0, 0};              // higher dims unused
    v4i g3 = {0, 0, 0, 0};
    v8i g4 = {0, 0, 0, 0, 0, 0, 0, 0};  // 6-arg toolchain variant: extra group

    __builtin_amdgcn_tensor_load_to_lds(g0, g1, g2, g3, g4, 0);
}

// Transpose + activate one (85 x 52) tile into 52*85 contiguous outputs.
__device__ __forceinline__ void decode_row(const float* __restrict__ t,
                                           float* __restrict__ dst,
                                           float fy, float aw, float ah) {
    for (int j = (int)threadIdx.x; j < TILE_ELEMS; j += 256) {
        const int xi   = j / NATTR;        // grid x (0..51)
        const int attr = j - xi * NATTR;   // attribute (0..84)
        const float v  = t[attr * GDIM + xi];
        float r;
        if (attr >= 4) {
            r = fast_sigmoid(v);                       // conf + classes
        } else if (attr == 0) {
            r = (fast_sigmoid(v) + (float)xi) * 8.0f;  // (sigmoid(tx)+grid_x)*stride
        } else if (attr == 1) {
            r = (fast_sigmoid(v) + fy) * 8.0f;         // (sigmoid(ty)+grid_y)*stride
        } else if (attr == 2) {
            r = __expf(v) * aw;                        // exp(tw)*anchor_w (stride folds out)
        } else {
            r = __expf(v) * ah;                        // exp(th)*anchor_h
        }
        __builtin_nontemporal_store(r, dst + j);       // streaming store, keep L2 clean
    }
}

__global__ __launch_bounds__(256) void yolo_head_kernel(const float* __restrict__ x,
                                                        float* __restrict__ out) {
    __shared__ float tile[ROWS][TILE_ELEMS];

    const int bid = blockIdx.x;
    const int yp  = bid % (GDIM / ROWS);
    const int a   = (bid / (GDIM / ROWS)) % NA;
    const int b   = bid / ((GDIM / ROWS) * NA);
    const int y0  = yp * ROWS;

    // Input tile start for row y: x[b][a*85][y][0]; rows at stride GDIM*GDIM
    const float* src0 = x + ((((u64)b * (NA * NATTR) + (u64)a * NATTR) * GDIM + (u64)y0) * GDIM);

    // ---- Issue both DMAs back-to-back (in-order completion on TENSORcnt) ----
    if (threadIdx.x < 32) {  // wave 0 only (uniform branch per wave32)
        tdm_load_tile(src0,        (u32)(uintptr_t)&tile[0][0]);
        tdm_load_tile(src0 + GDIM, (u32)(uintptr_t)&tile[1][0]);   // y0+1: +52 floats
        __builtin_amdgcn_s_wait_tensorcnt(1);   // tile 0 landed; tile 1 still in flight
    }
    __syncthreads();

    // Raw anchors (scaled_anchor * STRIDE == raw anchor, since STRIDE = 416/52 = 8)
    const float AW[NA] = {10.0f, 16.0f, 33.0f};
    const float AH[NA] = {13.0f, 30.0f, 23.0f};
    const float aw = AW[a];
    const float ah = AH[a];

    // Contiguous output chunk per (b, a, y): ((b*8112) + a*2704 + y*52) * 85
    float* __restrict__ dst0 =
        out + ((u64)b * (NA * GDIM * GDIM) + (u64)a * (GDIM * GDIM) + (u64)y0 * GDIM) * NATTR;

    decode_row(&tile[0][0], dst0, (float)y0, aw, ah);

    if (threadIdx.x < 32) {
        __builtin_amdgcn_s_wait_tensorcnt(0);   // tile 1 landed
    }
    __syncthreads();

    decode_row(&tile[1][0], dst0 + TILE_ELEMS, (float)(y0 + 1), aw, ah);
}

extern "C" void kernel_launch(void* const* d_in, const int* in_sizes, int n_in,
                              void* d_out, int out_size, void* d_ws, size_t ws_size,
                              hipStream_t stream) {
    (void)in_sizes; (void)n_in; (void)d_ws; (void)ws_size; (void)out_size;
    const float* x = (const float*)d_in[0];
    float* out = (float*)d_out;
    yolo_head_kernel<<<NBLOCKS, 256, 0, stream>>>(x, out);
}